// PhysicsInformedLoss_75076028334134
// MI455X (gfx1250) — compile-verified
//
#include <hip/hip_runtime.h>
#include <math.h>
#include <stdint.h>

typedef __attribute__((ext_vector_type(2))) float v2f;
typedef __attribute__((ext_vector_type(4))) float v4f;
typedef __attribute__((ext_vector_type(8))) float v8f;

#define B_  128
#define M_  35
#define P_  224
#define PP_ (P_ * P_)            // 50176
#define NBP (B_ * PP_)           // 6422528
#define EPSF 1e-8f
#define SIGN_PENALTY_F 10.0f
#define RECON_WEIGHT_F 0.4f
#define DEFOCUS_RAD_F 1.0f
#define PI_F 3.14159265358979323846f
#define APAD 228                 // padded LDS row stride (bank-conflict free)

// ---------------------------------------------------------------------------
// WMMA helper: D = A(16x4) * B(4x16) + C, f32.
// A frag (lane l): m=l&15, kg=l>>4; holds K=kg*2, kg*2+1.
// B frag (lane l): n=l&15, kg=l>>4; holds K=kg*2, kg*2+1.
// C/D (lane l, vgpr r): n=l&15, m=r+8*(l>>4).
// ---------------------------------------------------------------------------
__device__ __forceinline__ v8f wmma_f32(v2f a, v2f b, v8f c) {
  return __builtin_amdgcn_wmma_f32_16x16x4_f32(false, a, false, b, (short)0, c,
                                               false, false);
}

// ---------------------------------------------------------------------------
// CDNA5 async global->LDS copy (16B per lane), tracked by ASYNCcnt.
// ---------------------------------------------------------------------------
__device__ __forceinline__ void async_copy_f4(const float* g, float* l) {
  uint32_t lds = (uint32_t)(uintptr_t)l;            // low 32b = LDS byte addr
  uint64_t ga = (uint64_t)(uintptr_t)g;
  asm volatile("global_load_async_to_lds_b128 %0, %1, off"
               :
               : "v"(lds), "v"(ga)
               : "memory");
}
__device__ __forceinline__ void async_wait0() {
  asm volatile("s_wait_asynccnt 0x0" ::: "memory");
}

// ---------------------------------------------------------------------------
// K0: fftshift-folded DFT matrices. Fs[k][n] = exp(-2pi*i*((k+112)%224)*n/224)
// ---------------------------------------------------------------------------
__global__ void k_dft_init(float* __restrict__ FsR, float* __restrict__ FsI,
                           float* __restrict__ FsnI, float* __restrict__ FsTR,
                           float* __restrict__ FsTI, float* __restrict__ FsTnI) {
  int idx = blockIdx.x * blockDim.x + threadIdx.x;
  if (idx >= PP_) return;
  int k = idx / P_;
  int n = idx % P_;
  int kp = (k + P_ / 2) % P_;
  int t = (kp * n) % P_;                     // exact integer phase reduction
  float ang = -2.0f * PI_F * (float)t / (float)P_;
  float s, c;
  __sincosf(ang, &s, &c);
  FsR[k * P_ + n] = c;
  FsI[k * P_ + n] = s;
  FsnI[k * P_ + n] = -s;
  FsTR[n * P_ + k] = c;
  FsTI[n * P_ + k] = s;
  FsTnI[n * P_ + k] = -s;
}

__global__ void k_zero(float* __restrict__ acc, float* __restrict__ sums,
                       int zero_acc) {
  int i = threadIdx.x;
  if (zero_acc && i < 4) acc[i] = 0.0f;
  if (i < B_) sums[i] = 0.0f;
}

// ---------------------------------------------------------------------------
// K1: sign-weighted MSE on Zernike coefficients (sum into acc[0]).
// ---------------------------------------------------------------------------
__global__ void k_zloss(const float* __restrict__ pred,
                        const float* __restrict__ tgt, float* __restrict__ acc) {
  __shared__ float red[256];
  int i = blockIdx.x * 256 + threadIdx.x;
  float v = 0.0f;
  if (i < B_ * M_) {
    float p = pred[i], t = tgt[i];
    float d = p - t;
    float sp = (p > 0.0f) ? 1.0f : ((p < 0.0f) ? -1.0f : 0.0f);
    float st = (t > 0.0f) ? 1.0f : ((t < 0.0f) ? -1.0f : 0.0f);
    float w = (sp * st < 0.0f) ? SIGN_PENALTY_F : 1.0f;
    v = d * d * w;
  }
  red[threadIdx.x] = v;
  __syncthreads();
  for (int s = 128; s > 0; s >>= 1) {
    if (threadIdx.x < s) red[threadIdx.x] += red[threadIdx.x + s];
    __syncthreads();
  }
  if (threadIdx.x == 0) atomicAdd(&acc[0], red[0]);
}

// ---------------------------------------------------------------------------
// K2: phase = pred(128x35) @ zern(35x50176), masked. WMMA f32 16x16x4.
// ---------------------------------------------------------------------------
__global__ void k_phase_gemm(const float* __restrict__ pred,
                             const float* __restrict__ zern,
                             const float* __restrict__ mask,
                             float* __restrict__ phase) {
  int wave = threadIdx.x >> 5;
  int lane = threadIdx.x & 31;
  int colTile = blockIdx.x * (blockDim.x >> 5) + wave;  // 0..3135
  int rowTile = blockIdx.y;                             // 0..7
  int mrow = lane & 15;
  int kg = lane >> 4;
  int ncol = colTile * 16 + (lane & 15);
  int arow = (rowTile * 16 + mrow) * M_;

  v8f acc = {0.f, 0.f, 0.f, 0.f, 0.f, 0.f, 0.f, 0.f};

  for (int k0 = 0; k0 < 36; k0 += 4) {
    int ka = k0 + kg * 2;
    int ka0 = min(ka, M_ - 1);
    int ka1 = min(ka + 1, M_ - 1);
    float m0 = (ka < M_) ? 1.0f : 0.0f;
    float m1 = (ka + 1 < M_) ? 1.0f : 0.0f;
    v2f a, b;
    a.x = pred[arow + ka0] * m0;
    a.y = pred[arow + ka1] * m1;
    b.x = zern[ka0 * PP_ + ncol] * m0;
    b.y = zern[ka1 * PP_ + ncol] * m1;
    acc = wmma_f32(a, b, acc);
  }

  int mb = kg * 8;
  float mk = mask[ncol];
  for (int r = 0; r < 8; ++r) {
    int row = rowTile * 16 + mb + r;
    phase[row * PP_ + ncol] = acc[r] * mk;
  }
}

// ---------------------------------------------------------------------------
// K3a: pupil field E = mask * exp(i*(phase [+ defocus])), float4 streaming.
// ---------------------------------------------------------------------------
__global__ void k_pupil(const float* __restrict__ phase,
                        const float* __restrict__ mask, float* __restrict__ Er,
                        float* __restrict__ Ei, int variant) {
  int q = blockIdx.x * blockDim.x + threadIdx.x;   // float4 index
  if (q >= NBP / 4) return;
  int base = q * 4;
  int pix = base % PP_;
  v4f ph = *(const v4f*)(phase + base);
  v4f mk = *(const v4f*)(mask + pix);
  if (variant) {
    int iy = pix / P_;
    int ix = pix % P_;
    float ly = -1.0f + 2.0f * (float)iy / (float)(P_ - 1);
    float y2 = ly * ly;
    for (int j = 0; j < 4; ++j) {
      float lx = -1.0f + 2.0f * (float)(ix + j) / (float)(P_ - 1);
      float r2 = lx * lx + y2;
      ph[j] += DEFOCUS_RAD_F * (2.0f * r2 - 1.0f);
    }
  }
  v4f er, ei;
  for (int j = 0; j < 4; ++j) {
    float s, c;
    __sincosf(ph[j], &s, &c);
    er[j] = mk[j] * c;
    ei[j] = mk[j] * s;
  }
  *(v4f*)(Er + base) = er;
  *(v4f*)(Ei + base) = ei;
}

// ---------------------------------------------------------------------------
// K3b: row-DFT  Y = X @ Fs^T  (complex). grid=(224,14): blockIdx.y = ct.
// All 8 waves share ct => stage B-slab (3 x 224x16 f32) into LDS via
// CDNA5 async global->LDS, then feed WMMA B-fragments from LDS.
// ---------------------------------------------------------------------------
__global__ void k_fft_rows(const float* __restrict__ Er,
                           const float* __restrict__ Ei,
                           const float* __restrict__ FsTR,
                           const float* __restrict__ FsTI,
                           const float* __restrict__ FsTnI,
                           float* __restrict__ Yr, float* __restrict__ Yi) {
  __shared__ float BsR[P_ * 16];
  __shared__ float BsI[P_ * 16];
  __shared__ float BsnI[P_ * 16];

  int wave = threadIdx.x >> 5;
  int lane = threadIdx.x & 31;
  int ct = blockIdx.y;                                  // 0..13
  int w = blockIdx.x * (blockDim.x >> 5) + wave;        // 0..1791
  int b = w / 14;
  int rt = w % 14;

  // --- stage B slab: [k][n] for n in this ct, via async-to-LDS (b128/lane)
  {
    int nb = ct * 16;
    for (int f = threadIdx.x; f < 896; f += 256) {      // 224 rows * 4 quads
      int k = f >> 2;
      int n4 = (f & 3) * 4;
      async_copy_f4(&FsTR[k * P_ + nb + n4], &BsR[k * 16 + n4]);
      async_copy_f4(&FsTI[k * P_ + nb + n4], &BsI[k * 16 + n4]);
      async_copy_f4(&FsTnI[k * P_ + nb + n4], &BsnI[k * 16 + n4]);
    }
    async_wait0();
  }
  __syncthreads();

  int mrow = lane & 15;
  int kg = lane >> 4;
  int nn = lane & 15;
  int n16 = ct * 16 + nn;
  int arow = (rt * 16 + mrow) * P_;
  const float* er = Er + b * PP_;
  const float* ei = Ei + b * PP_;

  v8f ar = {0.f, 0.f, 0.f, 0.f, 0.f, 0.f, 0.f, 0.f};
  v8f ai = {0.f, 0.f, 0.f, 0.f, 0.f, 0.f, 0.f, 0.f};

  for (int k0 = 0; k0 < P_; k0 += 4) {
    int ka = k0 + kg * 2;
    v2f axr, axi, bR, bI, bnI;
    axr.x = er[arow + ka];
    axr.y = er[arow + ka + 1];
    axi.x = ei[arow + ka];
    axi.y = ei[arow + ka + 1];
    bR.x = BsR[ka * 16 + nn];
    bR.y = BsR[(ka + 1) * 16 + nn];
    bI.x = BsI[ka * 16 + nn];
    bI.y = BsI[(ka + 1) * 16 + nn];
    bnI.x = BsnI[ka * 16 + nn];
    bnI.y = BsnI[(ka + 1) * 16 + nn];
    __builtin_prefetch(&er[arow + k0 + 8], 0, 1);
    ar = wmma_f32(axr, bR, ar);
    ar = wmma_f32(axi, bnI, ar);
    ai = wmma_f32(axr, bI, ai);
    ai = wmma_f32(axi, bR, ai);
  }

  int mb = kg * 8;
  for (int r = 0; r < 8; ++r) {
    int row = rt * 16 + mb + r;
    Yr[b * PP_ + row * P_ + n16] = ar[r];
    Yi[b * PP_ + row * P_ + n16] = ai[r];
  }
}

// ---------------------------------------------------------------------------
// K4: col-DFT  Z = Fs @ Y, psf = |Z|^2, per-batch sums.
// grid=(224,14): blockIdx.y = rt. 8 waves share the Fs row-slab => stage
// A-slab (3 x 16x224, padded stride 228) into LDS via async-to-LDS.
// ---------------------------------------------------------------------------
__global__ void k_fft_cols(const float* __restrict__ Yr,
                           const float* __restrict__ Yi,
                           const float* __restrict__ FsR,
                           const float* __restrict__ FsI,
                           const float* __restrict__ FsnI,
                           float* __restrict__ psf, float* __restrict__ sums) {
  __shared__ float AsR[16 * APAD];
  __shared__ float AsI[16 * APAD];
  __shared__ float AsnI[16 * APAD];

  int wave = threadIdx.x >> 5;
  int lane = threadIdx.x & 31;
  int rt = blockIdx.y;                                  // 0..13
  int w = blockIdx.x * (blockDim.x >> 5) + wave;        // 0..1791
  int b = w / 14;
  int ct = w % 14;

  // --- stage A slab: Fs rows [rt*16 .. rt*16+15], padded stride APAD
  {
    int rbase = rt * 16;
    for (int f = threadIdx.x; f < 896; f += 256) {      // 16 rows * 56 quads
      int row = f / 56;
      int c4 = (f % 56) * 4;
      async_copy_f4(&FsR[(rbase + row) * P_ + c4], &AsR[row * APAD + c4]);
      async_copy_f4(&FsI[(rbase + row) * P_ + c4], &AsI[row * APAD + c4]);
      async_copy_f4(&FsnI[(rbase + row) * P_ + c4], &AsnI[row * APAD + c4]);
    }
    async_wait0();
  }
  __syncthreads();

  int mrow = lane & 15;
  int kg = lane >> 4;
  int n16 = ct * 16 + (lane & 15);
  const float* yr = Yr + b * PP_;
  const float* yi = Yi + b * PP_;

  v8f zr = {0.f, 0.f, 0.f, 0.f, 0.f, 0.f, 0.f, 0.f};
  v8f zi = {0.f, 0.f, 0.f, 0.f, 0.f, 0.f, 0.f, 0.f};

  for (int k0 = 0; k0 < P_; k0 += 4) {
    int ka = k0 + kg * 2;
    v2f aR, aI, anI, byr, byi;
    aR.x = AsR[mrow * APAD + ka];
    aR.y = AsR[mrow * APAD + ka + 1];
    aI.x = AsI[mrow * APAD + ka];
    aI.y = AsI[mrow * APAD + ka + 1];
    anI.x = AsnI[mrow * APAD + ka];
    anI.y = AsnI[mrow * APAD + ka + 1];
    byr.x = yr[ka * P_ + n16];
    byr.y = yr[(ka + 1) * P_ + n16];
    byi.x = yi[ka * P_ + n16];
    byi.y = yi[(ka + 1) * P_ + n16];
    __builtin_prefetch(&yr[(k0 + 4) * P_ + n16], 0, 1);
    zr = wmma_f32(aR, byr, zr);
    zr = wmma_f32(anI, byi, zr);
    zi = wmma_f32(aR, byi, zi);
    zi = wmma_f32(aI, byr, zi);
  }

  int mb = kg * 8;
  float local = 0.0f;
  for (int r = 0; r < 8; ++r) {
    float p = zr[r] * zr[r] + zi[r] * zi[r];
    psf[b * PP_ + (rt * 16 + mb + r) * P_ + n16] = p;
    local += p;
  }
  for (int off = 16; off > 0; off >>= 1) local += __shfl_xor(local, off, 32);
  if (lane == 0) atomicAdd(&sums[b], local);
}

// ---------------------------------------------------------------------------
// K5: recon MSE: sum (psf/(sum+eps) - target)^2 into acc[1]. float4 stream.
// ---------------------------------------------------------------------------
__global__ void k_recon_mse(const float* __restrict__ psf,
                            const float* __restrict__ sums,
                            const float* __restrict__ input_psfs, int variant,
                            float* __restrict__ acc) {
  __shared__ float red[256];
  int q = blockIdx.x * 256 + threadIdx.x;   // float4 index
  float v = 0.0f;
  if (q < NBP / 4) {
    int base = q * 4;
    int b = base / PP_;
    int pix = base % PP_;
    v4f pv = *(const v4f*)(psf + base);
    v4f tv = *(const v4f*)(input_psfs + (b * 2 + variant) * PP_ + pix);
    float inv = 1.0f / (sums[b] + EPSF);
    for (int j = 0; j < 4; ++j) {
      float d = pv[j] * inv - tv[j];
      v += d * d;
    }
  }
  red[threadIdx.x] = v;
  __syncthreads();
  for (int s = 128; s > 0; s >>= 1) {
    if (threadIdx.x < s) red[threadIdx.x] += red[threadIdx.x + s];
    __syncthreads();
  }
  if (threadIdx.x == 0) atomicAdd(&acc[1], red[0]);
}

__global__ void k_final(const float* __restrict__ acc, float* __restrict__ out) {
  if (threadIdx.x == 0) {
    float zl = acc[0] / (float)(B_ * M_);
    float rl = acc[1] / (float)NBP;
    out[0] = zl + RECON_WEIGHT_F * rl;
  }
}

// ---------------------------------------------------------------------------
extern "C" void kernel_launch(void* const* d_in, const int* in_sizes, int n_in,
                              void* d_out, int out_size, void* d_ws,
                              size_t ws_size, hipStream_t stream) {
  const float* pred = (const float*)d_in[0];        // [128,35]
  const float* target = (const float*)d_in[1];      // [128,35]
  const float* input_psfs = (const float*)d_in[2];  // [128,2,224,224]
  const float* zern = (const float*)d_in[3];        // [35,224,224]
  const float* mask = (const float*)d_in[4];        // [224,224]
  float* out = (float*)d_out;

  float* ws = (float*)d_ws;
  float* acc = ws;           // [0]=z_loss sum, [1]=recon sum
  float* sums = ws + 4;      // 128 per-batch psf sums
  float* FsR = ws + 256;
  float* FsI = FsR + PP_;
  float* FsnI = FsI + PP_;
  float* FsTR = FsnI + PP_;
  float* FsTI = FsTR + PP_;
  float* FsTnI = FsTI + PP_;
  float* phase = FsTnI + PP_;          // [128,224,224]
  float* Er = phase + (size_t)NBP;
  float* Ei = Er + (size_t)NBP;
  float* Yr = Ei + (size_t)NBP;
  float* Yi = Yr + (size_t)NBP;
  float* psf = Yi + (size_t)NBP;

  k_zero<<<1, 256, 0, stream>>>(acc, sums, 1);
  k_dft_init<<<(PP_ + 255) / 256, 256, 0, stream>>>(FsR, FsI, FsnI, FsTR, FsTI,
                                                    FsTnI);
  k_zloss<<<(B_ * M_ + 255) / 256, 256, 0, stream>>>(pred, target, acc);
  k_phase_gemm<<<dim3(3136 / 8, 8), 256, 0, stream>>>(pred, zern, mask, phase);

  for (int variant = 0; variant < 2; ++variant) {
    if (variant) k_zero<<<1, 256, 0, stream>>>(acc, sums, 0);
    k_pupil<<<(NBP / 4) / 256, 256, 0, stream>>>(phase, mask, Er, Ei, variant);
    k_fft_rows<<<dim3(224, 14), 256, 0, stream>>>(Er, Ei, FsTR, FsTI, FsTnI,
                                                  Yr, Yi);
    k_fft_cols<<<dim3(224, 14), 256, 0, stream>>>(Yr, Yi, FsR, FsI, FsnI, psf,
                                                  sums);
    k_recon_mse<<<(NBP / 4) / 256, 256, 0, stream>>>(psf, sums, input_psfs,
                                                     variant, acc);
  }

  k_final<<<1, 32, 0, stream>>>(acc, out);
}